// OPDMPConv_53017076301931
// MI455X (gfx1250) — compile-verified
//
#include <hip/hip_runtime.h>
#include <math.h>

#define NN   100000
#define EE   1000000
#define PP   3
#define INC  256
#define OUTC 128
#define FIN  512          // OUT_C * 4
#define LDA1 (INC + 4)    // padded LDS stride (words%64 = 4) -> conflict free
#define LDA2 (FIN + 4)    // padded LDS stride (words%64 = 4) -> conflict free

typedef float v2f __attribute__((ext_vector_type(2)));
typedef float v8f __attribute__((ext_vector_type(8)));

// Relaxed, agent-scope f32 atomic add -> selects native global_atomic_add_f32
// (no return slot used -> atomic-without-return, tracked on STOREcnt).
__device__ __forceinline__ void atomic_add_f32(float* addr, float v) {
  __hip_atomic_fetch_add(addr, v, __ATOMIC_RELAXED, __HIP_MEMORY_SCOPE_AGENT);
}

// ---------------------------------------------------------------- utilities
__global__ void zero4_kernel(float4* __restrict__ p, long n4) {
  long i = (long)blockIdx.x * blockDim.x + threadIdx.x;
  long stride = (long)gridDim.x * blockDim.x;
  float4 z = make_float4(0.f, 0.f, 0.f, 0.f);
  for (; i < n4; i += stride) p[i] = z;
}

__global__ __launch_bounds__(256) void deg_kernel(const int* __restrict__ row,
                                                  const float* __restrict__ w,
                                                  float* __restrict__ deg) {
  int e = blockIdx.x * 256 + threadIdx.x;
  if (e < EE) atomic_add_f32(&deg[row[e]], w[e]);
}

__global__ __launch_bounds__(256) void dinv_kernel(const float* __restrict__ deg,
                                                   float* __restrict__ dinv) {
  int i = blockIdx.x * 256 + threadIdx.x;
  if (i < NN) {
    float d = deg[i];
    dinv[i] = d > 0.f ? 1.f / d : 0.f;
  }
}

// -------------------------------------------------- GEMM1: x_trans = x@W + b
// block = 256 threads = 8 waves; block owns 16 rows, wave w owns cols [16w,16w+16)
__global__ __launch_bounds__(256) void gemm_lin_kernel(
    const float* __restrict__ x, const float* __restrict__ W,
    const float* __restrict__ b, float* __restrict__ xt) {
  __shared__ float As[16 * LDA1];
  const int rb = blockIdx.x * 16;
  const int t = threadIdx.x;
  // cooperative A-tile load: thread t loads 16 floats of row t/16
  {
    const int r = t >> 4;
    const int kc = (t & 15) << 4;
    const float4* src = (const float4*)(x + (long)(rb + r) * INC + kc);
    float4* dst = (float4*)(As + r * LDA1 + kc);
    dst[0] = src[0]; dst[1] = src[1]; dst[2] = src[2]; dst[3] = src[3];
  }
  __syncthreads();

  const int lane = t & 31;
  const int wave = t >> 5;
  const int half = lane >> 4;   // 0: K pair {k,k+1}; 1: K pair {k+2,k+3}
  const int lm   = lane & 15;
  const int colb = wave * 16;

  v8f c = {0.f, 0.f, 0.f, 0.f, 0.f, 0.f, 0.f, 0.f};
  const float* Arow = As + lm * LDA1;
  #pragma unroll 4
  for (int k0 = 0; k0 < INC; k0 += 4) {
    v2f a, bb;
    const int ka = k0 + 2 * half;
    a.x  = Arow[ka];
    a.y  = Arow[ka + 1];
    bb.x = W[(long)ka * OUTC + colb + lm];
    bb.y = W[(long)(ka + 1) * OUTC + colb + lm];
    c = __builtin_amdgcn_wmma_f32_16x16x4_f32(false, a, false, bb,
                                              (short)0, c, false, false);
  }

  const int col = colb + lm;
  const float bias = b[col];
  #pragma unroll
  for (int j = 0; j < 8; ++j) {
    const int m = half ? (j + 8) : j;
    xt[(long)(rb + m) * OUTC + col] = c[j] + bias;
  }
}

// ------------------------------------------------- edge scatter (one wave/edge)
__global__ __launch_bounds__(256) void scatter_kernel(
    const int* __restrict__ rows, const int* __restrict__ cols,
    const float* __restrict__ w, const float* __restrict__ pe,
    const float* __restrict__ xt, float* __restrict__ out_avg,
    float* __restrict__ out_dir) {
  const int e = blockIdx.x * 8 + (threadIdx.x >> 5);
  if (e >= EE) return;
  const int lane = threadIdx.x & 31;
  const int r = rows[e];
  const int cl = cols[e];
  const float we = w[e];
  const float diff = pe[cl] - pe[r];
  const float wu = diff > 0.f ? diff : 0.f;    // relu(diff)
  const float wd = diff < 0.f ? -diff : 0.f;   // relu(-diff)

  const float4 xv = *(const float4*)(xt + (long)cl * OUTC + lane * 4);
  float* pa = out_avg + (long)r * OUTC + lane * 4;
  float* pd = out_dir + (long)r * (2 * OUTC) + lane * 8;

  atomic_add_f32(pa + 0, we * xv.x);
  atomic_add_f32(pa + 1, we * xv.y);
  atomic_add_f32(pa + 2, we * xv.z);
  atomic_add_f32(pa + 3, we * xv.w);

  atomic_add_f32(pd + 0, wu * xv.x); atomic_add_f32(pd + 1, wd * xv.x);
  atomic_add_f32(pd + 2, wu * xv.y); atomic_add_f32(pd + 3, wd * xv.y);
  atomic_add_f32(pd + 4, wu * xv.z); atomic_add_f32(pd + 5, wd * xv.z);
  atomic_add_f32(pd + 6, wu * xv.w); atomic_add_f32(pd + 7, wd * xv.w);
}

// --------------------------- fusion GEMM + softmax weight + leaky_relu + accum
__global__ __launch_bounds__(256) void fusion_kernel(
    const float* __restrict__ xt, const float* __restrict__ out_avg,
    const float* __restrict__ out_dir, const float* __restrict__ dinv,
    const float* __restrict__ fusW, const float* __restrict__ fusb,
    const float* __restrict__ dmat, const float* __restrict__ hopb,
    float* __restrict__ out, int p) {
  __shared__ float As[16 * LDA2];
  const int rb = blockIdx.x * 16;
  const int t = threadIdx.x;
  // assemble A tile: [x_trans | out_avg*dinv | out_dir*dinv], 32 floats/thread
  {
    const int r = t >> 4;           // row 0..15
    const int chunk = t & 15;       // 32-float chunk 0..15
    const long row = rb + r;
    const float di = dinv[row];
    const int ks = chunk * 32;
    const float* src;
    float scale;
    if (chunk < 4)      { src = xt      + row * OUTC       + ks;         scale = 1.f; }
    else if (chunk < 8) { src = out_avg + row * OUTC       + (ks - 128); scale = di;  }
    else                { src = out_dir + row * (2 * OUTC) + (ks - 256); scale = di;  }
    float* dst = As + r * LDA2 + ks;
    #pragma unroll
    for (int i = 0; i < 32; i += 4) {
      float4 v = *(const float4*)(src + i);
      dst[i + 0] = v.x * scale;
      dst[i + 1] = v.y * scale;
      dst[i + 2] = v.z * scale;
      dst[i + 3] = v.w * scale;
    }
  }
  __syncthreads();

  const int lane = t & 31;
  const int wave = t >> 5;
  const int half = lane >> 4;
  const int lm   = lane & 15;
  const int colb = wave * 16;
  const float* Bp = fusW + (long)p * FIN * OUTC;

  v8f c = {0.f, 0.f, 0.f, 0.f, 0.f, 0.f, 0.f, 0.f};
  const float* Arow = As + lm * LDA2;
  #pragma unroll 4
  for (int k0 = 0; k0 < FIN; k0 += 4) {
    v2f a, bb;
    const int ka = k0 + 2 * half;
    a.x  = Arow[ka];
    a.y  = Arow[ka + 1];
    bb.x = Bp[(long)ka * OUTC + colb + lm];
    bb.y = Bp[(long)(ka + 1) * OUTC + colb + lm];
    c = __builtin_amdgcn_wmma_f32_16x16x4_f32(false, a, false, bb,
                                              (short)0, c, false, false);
  }

  const int col = colb + lm;
  // softmax over hops of d[:, col]
  const float e0 = __expf(dmat[col]);
  const float e1 = __expf(dmat[OUTC + col]);
  const float e2 = __expf(dmat[2 * OUTC + col]);
  const float ep = (p == 0) ? e0 : (p == 1) ? e1 : e2;
  const float dw = ep / (e0 + e1 + e2);
  const float fb = fusb[p * OUTC + col];
  const float hb = hopb[p * OUTC + col];
  #pragma unroll
  for (int j = 0; j < 8; ++j) {
    const int m = half ? (j + 8) : j;
    float h = c[j] + fb;
    h = h > 0.f ? h : 0.01f * h;            // leaky_relu
    const long oi = (long)(rb + m) * OUTC + col;
    out[oi] += h * dw + hb;
  }
}

// ------------------------------------------------------------------- launcher
extern "C" void kernel_launch(void* const* d_in, const int* in_sizes, int n_in,
                              void* d_out, int out_size, void* d_ws, size_t ws_size,
                              hipStream_t stream) {
  const float* x    = (const float*)d_in[0];
  const float* pe   = (const float*)d_in[1];
  const int*   ei   = (const int*)d_in[2];
  const float* ew   = (const float*)d_in[3];
  const float* linW = (const float*)d_in[4];
  const float* linb = (const float*)d_in[5];
  const float* dmat = (const float*)d_in[6];
  const float* hopb = (const float*)d_in[7];
  const float* fusW = (const float*)d_in[8];
  const float* fusb = (const float*)d_in[9];
  float* out = (float*)d_out;

  float* ws      = (float*)d_ws;
  float* xt      = ws;                            // N*128
  float* deg     = xt + (long)NN * OUTC;          // N      (zeroed region start)
  float* out_avg = deg + NN;                      // N*128
  float* out_dir = out_avg + (long)NN * OUTC;     // N*256
  float* dinv    = out_dir + (long)NN * 2 * OUTC; // N

  // zero output accumulator
  zero4_kernel<<<4096, 256, 0, stream>>>((float4*)out, (long)NN * OUTC / 4);
  // x_trans = x @ lin_W + lin_b
  gemm_lin_kernel<<<NN / 16, 256, 0, stream>>>(x, linW, linb, xt);

  for (int p = 0; p < PP; ++p) {
    const int* rows = ei + (long)p * 2 * EE;
    const int* cols = rows + EE;
    const float* wp = ew + (long)p * EE;
    // zero deg + out_avg + out_dir (contiguous: N*(1+128+256) floats)
    zero4_kernel<<<4096, 256, 0, stream>>>((float4*)deg,
                                           (long)NN * (1 + OUTC + 2 * OUTC) / 4);
    deg_kernel<<<(EE + 255) / 256, 256, 0, stream>>>(rows, wp, deg);
    dinv_kernel<<<(NN + 255) / 256, 256, 0, stream>>>(deg, dinv);
    scatter_kernel<<<EE / 8, 256, 0, stream>>>(rows, cols, wp, pe, xt,
                                               out_avg, out_dir);
    fusion_kernel<<<NN / 16, 256, 0, stream>>>(xt, out_avg, out_dir, dinv,
                                               fusW, fusb, dmat, hopb, out, p);
  }
}